// TAGNodeReg_23184233464166
// MI455X (gfx1250) — compile-verified
//
#include <hip/hip_runtime.h>
#include <hip/hip_bf16.h>

// ---------------------------------------------------------------------------
// TAGConv GNN forward for MI455X (gfx1250, wave32).
//   Sparse hops  : edge-parallel gather/scatter-add (L2-resident node table)
//   Dense mixes  : V_WMMA_F32_16X16X4_F32 tiles, out = b + sum_k h^k @ W_k
// ---------------------------------------------------------------------------

typedef __attribute__((ext_vector_type(2))) float v2f;
typedef __attribute__((ext_vector_type(8))) float v8f;

#define NEG_SLOPE 0.01f

// ---------------------------------------------------------------- utilities
__global__ void fill_zero_kernel(float* __restrict__ p, long n) {
    long i = (long)blockIdx.x * blockDim.x + threadIdx.x;
    long stride = (long)gridDim.x * blockDim.x;
    for (; i < n; i += stride) p[i] = 0.0f;
}

// deg[d] += 1 for every edge (symmetric gcn_norm with all-ones weights)
__global__ void degree_kernel(const int* __restrict__ dst, float* __restrict__ deg, int E) {
    int e = blockIdx.x * blockDim.x + threadIdx.x;
    if (e >= E) return;
    atomicAdd(&deg[dst[e]], 1.0f);
}

// in-place: deg -> dis = deg>0 ? rsqrt(max(deg,1)) : 0
__global__ void dis_kernel(float* __restrict__ deg, int N) {
    int n = blockIdx.x * blockDim.x + threadIdx.x;
    if (n >= N) return;
    float d = deg[n];
    deg[n] = (d > 0.0f) ? rsqrtf(fmaxf(d, 1.0f)) : 0.0f;
}

__global__ void norm_kernel(const int* __restrict__ src, const int* __restrict__ dst,
                            const float* __restrict__ dis, float* __restrict__ norm, int E) {
    int e = blockIdx.x * blockDim.x + threadIdx.x;
    if (e >= E) return;
    norm[e] = dis[src[e]] * dis[dst[e]];
}

// ------------------------------------------------- sparse hop: h_out = A_n h_in
// One thread per edge: h_out[dst] += norm[e] * h_in[src]   (F floats)
template <int F>
__global__ void scatter_hop_kernel(const float* __restrict__ hin, float* __restrict__ hout,
                                   const int* __restrict__ src, const int* __restrict__ dst,
                                   const float* __restrict__ norm, int E) {
    int e = blockIdx.x * blockDim.x + threadIdx.x;
    if (e >= E) return;
    int s = src[e];
    int d = dst[e];
    float w = norm[e];
    if (F == 16) {
        const float4* hs = (const float4*)(hin + (size_t)s * 16);
        float* ho = hout + (size_t)d * 16;
#pragma unroll
        for (int q = 0; q < 4; ++q) {
            float4 v = hs[q];
            atomicAdd(ho + q * 4 + 0, w * v.x);
            atomicAdd(ho + q * 4 + 1, w * v.y);
            atomicAdd(ho + q * 4 + 2, w * v.z);
            atomicAdd(ho + q * 4 + 3, w * v.w);
        }
    } else {
        const float* hs = hin + (size_t)s * F;
        float* ho = hout + (size_t)d * F;
#pragma unroll
        for (int f = 0; f < F; ++f) atomicAdd(ho + f, w * hs[f]);
    }
}

// ------------------------------------------------------------- WMMA GEMM mix
// Accumulate one hop's contribution:  acc += A(16 nodes x F) * Wk(F x 16)
// f32 WMMA operand layout (ISA 7.12.2, 16x16x4 f32):
//   A : lane m=lane&15 holds row m; half=lane>>4 selects K base 2*half;
//       A.x = h[m][f0+2*half], A.y = h[m][f0+2*half+1]
//   B : lane n=lane&15 holds col n; B.x = W[f0+2*half][n], B.y = W[f0+2*half+1][n]
//   C/D: lane n=lane&15, vgpr r holds row m = r + 8*half
// All loads are unconditional (EXEC all-ones, no exec-mask branches); the F==2
// padding into the K=4 contraction is done with register selects.
template <int F>
__device__ inline v8f wmma_accum_hop(v8f acc, const float* __restrict__ hk,
                                     const float* __restrict__ Wk,
                                     int rowA, int m, int half) {
    const float* arow = hk + (size_t)rowA * F;
    if (F % 4 == 0) {
#pragma unroll
        for (int f0 = 0; f0 < F; f0 += 4) {
            const int ka = f0 + half * 2;           // even, ka+1 <= F-1
            v2f A = *(const v2f*)(arow + ka);       // 8B-aligned pair load
            v2f B;
            B.x = Wk[ka * 16 + m];
            B.y = Wk[ka * 16 + 16 + m];
            acc = __builtin_amdgcn_wmma_f32_16x16x4_f32(
                false, A, false, B, (short)0, acc, false, false);
        }
    } else {  // F == 2: features occupy K=0,1; K=2,3 (half==1 slots) are zero
        v2f a = *(const v2f*)arow;                  // h[m][0], h[m][1]
        v2f A, B;
        const bool hi = (half != 0);
        A.x = hi ? 0.0f : a.x;
        A.y = hi ? 0.0f : a.y;
        float w0 = Wk[m];                           // W row 0, col m
        float w1 = Wk[16 + m];                      // W row 1, col m
        B.x = hi ? 0.0f : w0;
        B.y = hi ? 0.0f : w1;
        acc = __builtin_amdgcn_wmma_f32_16x16x4_f32(
            false, A, false, B, (short)0, acc, false, false);
    }
    return acc;
}

// out[N x 16] = bias + sum_{k=0..4} h^k[N x F] @ W[k]  (W: [5, F, 16] row-major)
// One wave per 16-node tile, 8 waves per block.
template <int F, bool RELU>
__global__ void tag_gemm_kernel(const float* __restrict__ h0, const float* __restrict__ h1,
                                const float* __restrict__ h2, const float* __restrict__ h3,
                                const float* __restrict__ h4,
                                const float* __restrict__ W, const float* __restrict__ bias,
                                float* __restrict__ out, int N) {
    int wave = threadIdx.x >> 5;
    int lane = threadIdx.x & 31;
    int tile = blockIdx.x * (blockDim.x >> 5) + wave;
    int ntiles = (N + 15) >> 4;
    if (tile >= ntiles) return;  // wave-uniform: EXEC stays all-ones for WMMA

    int m = lane & 15;           // A row within tile / B,C column index
    int half = (lane >> 4) & 1;
    int row0 = tile * 16;
    int rowA = row0 + m;
    if (rowA >= N) rowA = N - 1; // clamp (keeps EXEC full; store is guarded)

    v8f acc;
    float bn = bias[m];          // column n = lane&15
#pragma unroll
    for (int r = 0; r < 8; ++r) acc[r] = bn;

    acc = wmma_accum_hop<F>(acc, h0, W + 0 * F * 16, rowA, m, half);
    acc = wmma_accum_hop<F>(acc, h1, W + 1 * F * 16, rowA, m, half);
    acc = wmma_accum_hop<F>(acc, h2, W + 2 * F * 16, rowA, m, half);
    acc = wmma_accum_hop<F>(acc, h3, W + 3 * F * 16, rowA, m, half);
    acc = wmma_accum_hop<F>(acc, h4, W + 4 * F * 16, rowA, m, half);

    if (RELU) {
#pragma unroll
        for (int r = 0; r < 8; ++r) {
            float v = acc[r];
            acc[r] = (v > 0.0f) ? v : NEG_SLOPE * v;
        }
    }

    // D layout: vgpr r -> row m = r + 8*half, col n = lane&15.
    float* orow = out + (size_t)(row0 + 8 * half) * 16 + m;
    if (row0 + 16 <= N) {
        // Full tile (the only case when N % 16 == 0): unguarded store clause.
#pragma unroll
        for (int r = 0; r < 8; ++r) orow[(size_t)r * 16] = acc[r];
    } else {
        // Ragged tail tile: per-row guard.
#pragma unroll
        for (int r = 0; r < 8; ++r) {
            if (row0 + 8 * half + r < N) orow[(size_t)r * 16] = acc[r];
        }
    }
}

// ------------------------------------------------------------- output head
// out[n] = dot(h[n, 0:16], Wout[0:16]) + bout
__global__ void head_kernel(const float* __restrict__ h, const float* __restrict__ Wout,
                            const float* __restrict__ bout, float* __restrict__ out, int N) {
    int n = blockIdx.x * blockDim.x + threadIdx.x;
    if (n >= N) return;
    const float4* hp = (const float4*)(h + (size_t)n * 16);
    const float4* wp = (const float4*)Wout;
    float s = 0.0f;
#pragma unroll
    for (int q = 0; q < 4; ++q) {
        float4 a = hp[q];
        float4 b = wp[q];
        s += a.x * b.x + a.y * b.y + a.z * b.z + a.w * b.w;
    }
    out[n] = s + bout[0];
}

// ---------------------------------------------------------------------------
extern "C" void kernel_launch(void* const* d_in, const int* in_sizes, int n_in,
                              void* d_out, int out_size, void* d_ws, size_t ws_size,
                              hipStream_t stream) {
    (void)n_in; (void)out_size; (void)ws_size;

    const float* x    = (const float*)d_in[0];   // [N, 2]
    const int*   ei   = (const int*)d_in[1];     // [2, E]
    /* d_in[2] = edge_attr, unused by the reference path */
    const float* W1   = (const float*)d_in[3];   // [5, 2, 16]
    const float* b1   = (const float*)d_in[4];   // [16]
    const float* W2   = (const float*)d_in[5];   // [5, 16, 16]
    const float* b2   = (const float*)d_in[6];
    const float* W3   = (const float*)d_in[7];
    const float* b3   = (const float*)d_in[8];
    const float* Wout = (const float*)d_in[9];   // [16, 1]
    const float* bout = (const float*)d_in[10];  // [1]
    float* out = (float*)d_out;                  // [N, 1]

    const int N = in_sizes[0] / 2;
    const int E = in_sizes[1] / 2;
    const int* src = ei;
    const int* dst = ei + E;

    // ---- workspace carve-up (256B aligned) -------------------------------
    char* base = (char*)d_ws;
    size_t off = 0;
    auto carve = [&](size_t bytes) -> float* {
        float* p = (float*)(base + off);
        off += (bytes + 255) & ~(size_t)255;
        return p;
    };
    float* norm = carve((size_t)E * 4);            // [E]
    float* deg  = carve((size_t)N * 4);            // [N] -> becomes dis in place
    float* hop[4];
    for (int k = 0; k < 4; ++k) hop[k] = carve((size_t)N * 16 * 4);
    float* actA = carve((size_t)N * 16 * 4);
    float* actB = carve((size_t)N * 16 * 4);

    const int BT = 256;
    const int gN  = (N + BT - 1) / BT;
    const int gE  = (E + BT - 1) / BT;
    const int ntiles = (N + 15) / 16;
    const int gT  = (ntiles + 7) / 8;  // 8 waves (16-node tiles) per 256-thread block

    // ---- gcn_norm: deg -> dis -> norm ------------------------------------
    fill_zero_kernel<<<1024, BT, 0, stream>>>(deg, (long)N);
    degree_kernel<<<gE, BT, 0, stream>>>(dst, deg, E);
    dis_kernel<<<gN, BT, 0, stream>>>(deg, N);
    norm_kernel<<<gE, BT, 0, stream>>>(src, dst, deg, norm, E);

    // ---- layer 1 (F=2), LeakyReLU ----------------------------------------
    {
        const float* hprev = x;
        for (int k = 0; k < 4; ++k) {
            fill_zero_kernel<<<1024, BT, 0, stream>>>(hop[k], (long)N * 2);
            scatter_hop_kernel<2><<<gE, BT, 0, stream>>>(hprev, hop[k], src, dst, norm, E);
            hprev = hop[k];
        }
        tag_gemm_kernel<2, true><<<gT, BT, 0, stream>>>(
            x, hop[0], hop[1], hop[2], hop[3], W1, b1, actA, N);
    }

    // ---- layer 2 (F=16), LeakyReLU ---------------------------------------
    {
        const float* hprev = actA;
        for (int k = 0; k < 4; ++k) {
            fill_zero_kernel<<<1024, BT, 0, stream>>>(hop[k], (long)N * 16);
            scatter_hop_kernel<16><<<gE, BT, 0, stream>>>(hprev, hop[k], src, dst, norm, E);
            hprev = hop[k];
        }
        tag_gemm_kernel<16, true><<<gT, BT, 0, stream>>>(
            actA, hop[0], hop[1], hop[2], hop[3], W2, b2, actB, N);
    }

    // ---- layer 3 (F=16), no activation -----------------------------------
    {
        const float* hprev = actB;
        for (int k = 0; k < 4; ++k) {
            fill_zero_kernel<<<1024, BT, 0, stream>>>(hop[k], (long)N * 16);
            scatter_hop_kernel<16><<<gE, BT, 0, stream>>>(hprev, hop[k], src, dst, norm, E);
            hprev = hop[k];
        }
        tag_gemm_kernel<16, false><<<gT, BT, 0, stream>>>(
            actB, hop[0], hop[1], hop[2], hop[3], W3, b3, actA, N);
    }

    // ---- regression head -------------------------------------------------
    head_kernel<<<gN, BT, 0, stream>>>(actA, Wout, bout, out, N);
}